// PatchSampleNonlocalOneGroup_89189290868990
// MI455X (gfx1250) — compile-verified
//
#include <hip/hip_runtime.h>
#include <math.h>

// ---------------------------------------------------------------------------
// Problem constants (fixed by the reference):
//  feat0: (8, 64,256,256)  feat1: (8,128,128,128)  feat2: (8,256,64,64)
//  search grid: h,w in [108,148)  (40x40 = 1600 candidates), patch 4x4, C=64
//  sample patch: rows/cols 126..129
//  outputs: x0 (8,256,64) | x1 (8,256,128) | x2 (8,256,256) | ids (8,256,2)
// ---------------------------------------------------------------------------

typedef __attribute__((ext_vector_type(2))) float v2f;
typedef __attribute__((ext_vector_type(8))) float v8f;

#define NB        8
#define NCAND     1600
#define GRID_W    40
#define LOC_BASE  108
#define SAMP_BASE 126

#define OUT0_OFF  0
#define OUT1_OFF  (8 * 256 * 64)                 // 131072
#define OUT2_OFF  (OUT1_OFF + 8 * 256 * 128)     // 393216
#define OUTI_OFF  (OUT2_OFF + 8 * 256 * 256)     // 917504

#define WS_DIST_FLOATS (NB * NCAND)              // 12800 floats
#define WS_IDS_OFF_BYTES (WS_DIST_FLOATS * 4)    // 51200 bytes (64B aligned)

// ---------------------------------------------------------------------------
// Kernel 1: squared L2 distance of each candidate patch vs center patch.
// One wave (32 lanes) per 16-candidate tile; K = 1024 consumed in chunks of 4
// via V_WMMA_F32_16X16X4_F32 with B = ones (row-sum through the matrix pipe).
// Two accumulators interleave to cover WMMA->WMMA RAW latency.
// ---------------------------------------------------------------------------
__global__ __launch_bounds__(32) void k_dist_wmma(const float* __restrict__ feat0,
                                                  float* __restrict__ dist)
{
    const int lane  = threadIdx.x;            // 0..31
    const int tile  = blockIdx.x;             // 0..99
    const int b     = blockIdx.y;             // 0..7
    const int m     = tile * 16 + (lane & 15);        // candidate id 0..1599
    const int ih    = m / GRID_W;
    const int iw    = m - ih * GRID_W;
    const int h     = LOC_BASE + ih;                  // candidate center row
    const int w     = LOC_BASE + iw;                  // candidate center col
    const int pxb   = (lane >> 4) << 1;               // px base: 0 (lanes<16) or 2

    const float* fb = feat0 + (size_t)b * 64 * 256 * 256;

    v2f ones; ones.x = 1.0f; ones.y = 1.0f;
    v8f acc0 = {};
    v8f acc1 = {};

    // 256 chunks: chunk q -> (channel c = q>>2, patch row py = q&3)
    for (int q = 0; q < 256; q += 2) {
        {
            const int c  = q >> 2;
            const int py = q & 3;
            const float* prow = fb + ((size_t)c * 256 + (h - 2 + py)) * 256 + (w - 2) + pxb;
            const float* srow = fb + ((size_t)c * 256 + (SAMP_BASE + py)) * 256 + SAMP_BASE + pxb;
            float d0 = prow[0] - srow[0];
            float d1 = prow[1] - srow[1];
            v2f a; a.x = d0 * d0; a.y = d1 * d1;
            acc0 = __builtin_amdgcn_wmma_f32_16x16x4_f32(false, a, false, ones,
                                                         (short)0, acc0, false, false);
        }
        {
            const int c  = (q + 1) >> 2;
            const int py = (q + 1) & 3;
            const float* prow = fb + ((size_t)c * 256 + (h - 2 + py)) * 256 + (w - 2) + pxb;
            const float* srow = fb + ((size_t)c * 256 + (SAMP_BASE + py)) * 256 + SAMP_BASE + pxb;
            float d0 = prow[0] - srow[0];
            float d1 = prow[1] - srow[1];
            v2f a; a.x = d0 * d0; a.y = d1 * d1;
            acc1 = __builtin_amdgcn_wmma_f32_16x16x4_f32(false, a, false, ones,
                                                         (short)0, acc1, false, false);
        }
    }
    v8f acc = acc0 + acc1;

    // D layout: VGPR r holds M=r (lanes 0-15) / M=8+r (lanes 16-31); all N equal.
    if (lane == 0 || lane == 16) {
        const int mbase = tile * 16 + ((lane == 16) ? 8 : 0);
        float* dp = dist + b * NCAND + mbase;
        dp[0] = acc[0]; dp[1] = acc[1]; dp[2] = acc[2]; dp[3] = acc[3];
        dp[4] = acc[4]; dp[5] = acc[5]; dp[6] = acc[6]; dp[7] = acc[7];
    }
}

// ---------------------------------------------------------------------------
// Kernel 2: exact top-256-smallest per batch via rank counting in LDS.
// rank(i) = #{ j : d[j] < d[i]  ||  (d[j]==d[i] && j < i) }  -> permutation,
// identical ordering to jax.lax.top_k(-d) (ascending d, ties by lower index).
// ---------------------------------------------------------------------------
__global__ __launch_bounds__(256) void k_topk(const float* __restrict__ dist,
                                              int* __restrict__ ids,
                                              float* __restrict__ out_ids)
{
    __shared__ float sd[NCAND];
    const int b   = blockIdx.x;
    const int tid = threadIdx.x;

    for (int i = tid; i < NCAND; i += 256) sd[i] = dist[b * NCAND + i];
    __syncthreads();

    for (int i = tid; i < NCAND; i += 256) {
        const float di = sd[i];
        int rank = 0;
        for (int j = 0; j < NCAND; ++j) {
            const float dj = sd[j];
            rank += (dj < di) || (dj == di && j < i);
        }
        if (rank < 256) {
            const int lh = LOC_BASE + i / GRID_W;
            const int lw = LOC_BASE + i % GRID_W;
            const int o  = (b * 256 + rank) * 2;
            ids[o]     = lh;
            ids[o + 1] = lw;
            out_ids[o]     = (float)lh;
            out_ids[o + 1] = (float)lw;
        }
    }
}

// ---------------------------------------------------------------------------
// Kernel 3: gather C-vector at scaled location and L2-normalize.
// One wave per (patch, batch, feature). fh = lh*H/256 == lh >> shift.
// ---------------------------------------------------------------------------
__global__ __launch_bounds__(32) void k_sample(const float* __restrict__ f0,
                                               const float* __restrict__ f1,
                                               const float* __restrict__ f2,
                                               const int* __restrict__ ids,
                                               float* __restrict__ out)
{
    const int p    = blockIdx.x;   // 0..255
    const int b    = blockIdx.y;   // 0..7
    const int fi   = blockIdx.z;   // 0..2
    const int lane = threadIdx.x;  // 0..31

    int C, H, shift;
    const float* f;
    size_t obase;
    if (fi == 0)      { C = 64;  H = 256; shift = 0; f = f0; obase = OUT0_OFF; }
    else if (fi == 1) { C = 128; H = 128; shift = 1; f = f1; obase = OUT1_OFF; }
    else              { C = 256; H = 64;  shift = 2; f = f2; obase = OUT2_OFF; }
    const int W = H;

    const int lh = ids[(b * 256 + p) * 2];
    const int lw = ids[(b * 256 + p) * 2 + 1];
    const int fh = lh >> shift;
    const int fw = lw >> shift;

    float v[8];
    float ss = 0.0f;
    const int iters = C >> 5;              // 2 / 4 / 8
    for (int t = 0; t < iters; ++t) {
        const int c = lane + (t << 5);
        const float x = f[(((size_t)b * C + c) * H + fh) * W + fw];
        v[t] = x;
        ss += x * x;
    }
    // wave32 butterfly reduction
    for (int off = 16; off >= 1; off >>= 1)
        ss += __shfl_xor(ss, off, 32);

    const float inv = 1.0f / (sqrtf(ss) + 1e-7f);
    float* o = out + obase + (size_t)(b * 256 + p) * C;
    for (int t = 0; t < iters; ++t)
        o[lane + (t << 5)] = v[t] * inv;
}

// ---------------------------------------------------------------------------
extern "C" void kernel_launch(void* const* d_in, const int* in_sizes, int n_in,
                              void* d_out, int out_size, void* d_ws, size_t ws_size,
                              hipStream_t stream)
{
    (void)in_sizes; (void)n_in; (void)out_size; (void)ws_size;

    const float* feat0 = (const float*)d_in[0];
    const float* feat1 = (const float*)d_in[1];
    const float* feat2 = (const float*)d_in[2];
    // d_in[3] = num_patches scalar (fixed at 256 by the reference setup)

    float* dist = (float*)d_ws;                                      // 8*1600 f32
    int*   ids  = (int*)((char*)d_ws + WS_IDS_OFF_BYTES);            // 8*256*2 i32
    float* out  = (float*)d_out;

    k_dist_wmma<<<dim3(100, 8), 32, 0, stream>>>(feat0, dist);
    k_topk   <<<dim3(8),       256, 0, stream>>>(dist, ids, out + OUTI_OFF);
    k_sample <<<dim3(256, 8, 3), 32, 0, stream>>>(feat0, feat1, feat2, ids, out);
}